// PairPosMLP_8546984919445
// MI455X (gfx1250) — compile-verified
//
#include <hip/hip_runtime.h>

typedef __attribute__((ext_vector_type(2))) float v2f;
typedef __attribute__((ext_vector_type(8))) float v8f;

constexpr int Bsz = 8, N1 = 1024, N2 = 1024, DIM = 256, HID = 16;
constexpr int WCOLS = 6 * HID + 1;   // 97

// ---------------------------------------------------------------------------
// Kernel 1: weights[B*N1, 97] = queries[B*N1, 256] @ Wg[256, 97] + bg
// One wave per 16x16 output tile; K=256 accumulated with 64 WMMA_F32_16X16X4.
// A layout: lane = M row (query row), VGPR0/1 = K, lanes>=16 take K+2.
// B layout: lane = N col (output col), VGPR0/1 = K, lanes>=16 take K+2.
// ---------------------------------------------------------------------------
__global__ __launch_bounds__(256) void weights_gemm_wmma(
        const float* __restrict__ q, const float* __restrict__ Wg,
        const float* __restrict__ bg, float* __restrict__ wout) {
    const int lane = threadIdx.x & 31;
    const int wid  = threadIdx.x >> 5;
    const int tile = blockIdx.x * 8 + wid;          // 512 row-tiles * 7 col-tiles
    const int COLT = 7;
    const int row0 = (tile / COLT) * 16;
    const int col0 = (tile % COLT) * 16;
    const int h    = lane & 15;
    const bool hiH = lane >= 16;
    const int koff = hiH ? 2 : 0;
    const int col  = col0 + h;
    const int colc = (col < WCOLS) ? col : (WCOLS - 1);   // clamp ragged edge

    const float* qrow = q + (size_t)(row0 + h) * DIM + koff;
    const float* wcol = Wg + colc;

    v8f c = {};
    #pragma unroll 4
    for (int kk = 0; kk < DIM; kk += 4) {
        v2f a, bm;
        a.x  = qrow[kk];
        a.y  = qrow[kk + 1];
        bm.x = wcol[(size_t)(kk + koff)     * WCOLS];
        bm.y = wcol[(size_t)(kk + koff + 1) * WCOLS];
        c = __builtin_amdgcn_wmma_f32_16x16x4_f32(false, a, false, bm,
                                                  (short)0, c, false, false);
    }

    const float bgv = bg[colc];
    const int rbase = row0 + (hiH ? 8 : 0);
    #pragma unroll
    for (int r = 0; r < 8; ++r) {
        if (col < WCOLS)
            wout[(size_t)(rbase + r) * WCOLS + col] = c[r] + bgv;
    }
}

// ---------------------------------------------------------------------------
// Kernel 2: per (b,m) wave, sweep n in 16-wide tiles.
//   A (held all loop) = w1^T : 16(h) x 4(k), lane = h, K split across halves.
//   B (per tile)      = feat^T: 4(k) x 16(n), lane = n; lanes 0-15 make dx,dy
//                       (k=0,1), lanes 16-31 make dw,dh (k=2,3).
//   D = hid^T (h x n): lane L VGPR r = hid[h = r + 8*(L>=16), n = L%16].
//   Layer 2: relu+fma per VGPR, one shfl_xor(16) to join h-halves,
//   lanes 0-15 store a coalesced 16-float output row.
// ---------------------------------------------------------------------------
__global__ __launch_bounds__(256) void pair_mlp_wmma(
        const float* __restrict__ pos1, const float* __restrict__ pos2,
        const float* __restrict__ wgt, float* __restrict__ out) {
    const int lane = threadIdx.x & 31;
    const int wid  = threadIdx.x >> 5;
    const int gid  = blockIdx.x * 8 + wid;          // (b,m) in [0, 8192)
    const int b    = gid >> 10;
    const bool hiH = lane >= 16;
    const int h    = lane & 15;

    // pos1 (wave-uniform)
    const float* p1 = pos1 + (size_t)gid * 4;
    const float x1  = p1[0], y1 = p1[1], w1p = p1[2], h1p = p1[3];
    const float c0  = hiH ? __logf(w1p) : x1;
    const float c1  = hiH ? __logf(h1p) : y1;
    const float s0  = hiH ? 1.0f : 1.0f / w1p;
    const float s1  = hiH ? 1.0f : 1.0f / h1p;

    // per-(b,m) generated weights
    const float* wrow = wgt + (size_t)gid * WCOLS;
    const int k0 = hiH ? 2 : 0;
    v2f a;                                  // A = w1^T, resident across tiles
    a.x = wrow[(k0 + 0) * HID + h];
    a.y = wrow[(k0 + 1) * HID + h];
    const int hb = hiH ? 8 : 0;
    float b1v[8], w2v[8];
    #pragma unroll
    for (int r = 0; r < 8; ++r) {
        b1v[r] = wrow[4 * HID + hb + r];
        w2v[r] = wrow[5 * HID + hb + r];
    }
    const float b2 = wrow[6 * HID];

    const float2* p2row = (const float2*)pos2 + (size_t)b * N2 * 2;
    float* orow = out + (size_t)gid * N2;

    for (int t = 0; t < N2 / 16; ++t) {
        const int n0 = t * 16;
        // lanes 0-15: (x2,y2); lanes 16-31: (w2,h2)
        const float2 p = p2row[(size_t)(n0 + h) * 2 + (hiH ? 1 : 0)];
        const float u0 = hiH ? __logf(p.x) : p.x;
        const float u1 = hiH ? __logf(p.y) : p.y;
        v2f f;
        f.x = (u0 - c0) * s0;
        f.y = (u1 - c1) * s1;

        v8f d = {};
        d = __builtin_amdgcn_wmma_f32_16x16x4_f32(false, a, false, f,
                                                  (short)0, d, false, false);
        float acc = 0.0f;
        #pragma unroll
        for (int r = 0; r < 8; ++r) {
            float tt = fmaxf(d[r] + b1v[r], 0.0f);
            acc = fmaf(tt, w2v[r], acc);
        }
        acc += __shfl_xor(acc, 16, 32);     // join h=0..7 and h=8..15 halves
        if (!hiH) orow[n0 + h] = acc + b2;  // coalesced 64B row store
    }
}

extern "C" void kernel_launch(void* const* d_in, const int* in_sizes, int n_in,
                              void* d_out, int out_size, void* d_ws, size_t ws_size,
                              hipStream_t stream) {
    const float* pos1    = (const float*)d_in[0];   // (8,1024,4)
    const float* pos2    = (const float*)d_in[1];   // (8,1024,4)
    const float* queries = (const float*)d_in[2];   // (8,1024,256)
    const float* Wg      = (const float*)d_in[3];   // (256,97)
    const float* bg      = (const float*)d_in[4];   // (97,)
    // d_in[5] = hidden_dim scalar (fixed 16, compile-time)

    float* wws = (float*)d_ws;                      // 8192 x 97 f32 ~ 3.03 MB

    // 512 row-tiles * 7 col-tiles = 3584 waves = 448 blocks of 8 waves
    weights_gemm_wmma<<<448, 256, 0, stream>>>(queries, Wg, bg, wws);
    // 8192 (b,m) waves = 1024 blocks of 8 waves
    pair_mlp_wmma<<<1024, 256, 0, stream>>>(pos1, pos2, wws, (float*)d_out);
}